// _ProposalTargetLayer_58978490909097
// MI455X (gfx1250) — compile-verified
//
#include <hip/hip_runtime.h>
#include <hip/hip_bf16.h>
#include <stdint.h>

// ---------------- problem constants (match reference) ----------------
#define NPROP   8000
#define GMAX    20
#define NTOT    (NPROP + GMAX)     // 8020 rois per image (gt appended)
#define HALF_N  (NTOT / 2)         // 4010, threefry counter split
#define RPI     128                // ROIS_PER_IMAGE
#define FGPI    32                 // FG_PER_IMAGE
#define NCLS    21
#define TW      (4 * NCLS)         // 84

#define AS1 __attribute__((address_space(1)))
#define AS3 __attribute__((address_space(3)))

#if defined(__has_builtin)
#if __has_builtin(__builtin_amdgcn_global_load_async_to_lds_b32)
#define HAVE_ASYNC_LDS 1
#endif
#endif

// ---------------- threefry2x32 (20 rounds), matches jax PRNG core ----
__device__ __forceinline__ unsigned rotl32(unsigned x, int d) {
  return (x << d) | (x >> (32 - d));
}

__device__ __forceinline__ void threefry2x32(unsigned k0, unsigned k1,
                                             unsigned x0, unsigned x1,
                                             unsigned& o0, unsigned& o1) {
  const unsigned ks2 = k0 ^ k1 ^ 0x1BD11BDAu;
  x0 += k0; x1 += k1;
#define TF_R(r) { x0 += x1; x1 = rotl32(x1, (r)); x1 ^= x0; }
  TF_R(13) TF_R(15) TF_R(26) TF_R(6)
  x0 += k1;  x1 += ks2 + 1u;
  TF_R(17) TF_R(29) TF_R(16) TF_R(24)
  x0 += ks2; x1 += k0 + 2u;
  TF_R(13) TF_R(15) TF_R(26) TF_R(6)
  x0 += k0;  x1 += k1 + 3u;
  TF_R(17) TF_R(29) TF_R(16) TF_R(24)
  x0 += k1;  x1 += ks2 + 4u;
  TF_R(13) TF_R(15) TF_R(26) TF_R(6)
  x0 += ks2; x1 += k0 + 5u;
#undef TF_R
  o0 = x0; o1 = x1;
}

// per-image key: jax.random.split(key(42), 128) counter layout
__device__ __forceinline__ void image_key(int img, unsigned& kk0, unsigned& kk1) {
  unsigned a0, a1;
  if (img < 64) {
    threefry2x32(0u, 42u, (unsigned)(2 * img),     (unsigned)(128 + 2 * img), a0, a1); kk0 = a0;
    threefry2x32(0u, 42u, (unsigned)(2 * img + 1), (unsigned)(129 + 2 * img), a0, a1); kk1 = a0;
  } else {
    threefry2x32(0u, 42u, (unsigned)(2 * img - 128), (unsigned)(2 * img),     a0, a1); kk0 = a1;
    threefry2x32(0u, 42u, (unsigned)(2 * img - 127), (unsigned)(2 * img + 1), a0, a1); kk1 = a1;
  }
}

// raw uniform bits for roi i, matching jax uniform counter split for N=8020
__device__ __forceinline__ unsigned rand_bits(unsigned kk0, unsigned kk1, int i) {
  unsigned a0, a1;
  if (i < HALF_N) { threefry2x32(kk0, kk1, (unsigned)i, (unsigned)(HALF_N + i), a0, a1); return a0; }
  threefry2x32(kk0, kk1, (unsigned)(i - HALF_N), (unsigned)i, a0, a1);
  return a1;
}

// ---------------- fused proposal-target kernel: 1 block per image ----
__global__ __launch_bounds__(256)
void proposal_target_kernel(const float* __restrict__ all_rois,
                            const float* __restrict__ gt_boxes,
                            const int*   __restrict__ num_boxes,
                            float* __restrict__ out, int B) {
  __shared__ float          s_gt[GMAX * 5];     // staged gt boxes (x1,y1,x2,y2,cls)
  __shared__ unsigned       s_key[NTOT];        // (rand19<<13)|idx
  __shared__ unsigned char  s_ab[NTOT];         // (class<<6)|gt_assign
  __shared__ unsigned       s_red[256];
  __shared__ int            s_keep[RPI];
  __shared__ unsigned char  s_isfg[RPI];
  __shared__ int            s_nfg;

  (void)num_boxes;  // reference ignores it (masks padded gt via zero-box check)

  const int img = blockIdx.x;
  const int t   = threadIdx.x;
  if (t == 0) s_nfg = 0;

  // ---- Phase 0: async-stage gt boxes (100 floats) into LDS -----------
  const float* gsrc = gt_boxes + (size_t)img * GMAX * 5;
#if defined(HAVE_ASYNC_LDS)
  if (t < GMAX * 5) {
    __builtin_amdgcn_global_load_async_to_lds_b32(
        (AS1 int*)(gsrc + t), (AS3 int*)(&s_gt[t]), 0, 0);
  }
#if __has_builtin(__builtin_amdgcn_s_wait_asynccnt)
  __builtin_amdgcn_s_wait_asynccnt(0);
#else
  asm volatile("s_wait_asynccnt 0" ::: "memory");
#endif
#else
  if (t < GMAX * 5) s_gt[t] = gsrc[t];
#endif
  __syncthreads();

  unsigned kk0, kk1;
  image_key(img, kk0, kk1);

  // ---- Phase A: IoU max/argmax + rand + classify for all 8020 rois ---
  int localfg = 0;
  for (int i = t; i < NTOT; i += 256) {
    float x1, y1, x2, y2;
    if (i < NPROP) {
      const float* rp = all_rois + ((size_t)img * NPROP + i) * 5;
      if (i + 256 < NPROP) __builtin_prefetch(rp + 5 * 256, 0, 0);
      x1 = rp[1]; y1 = rp[2]; x2 = rp[3]; y2 = rp[4];
    } else {
      const float* gp = &s_gt[(i - NPROP) * 5];
      x1 = gp[0]; y1 = gp[1]; x2 = gp[2]; y2 = gp[3];
    }
    const float bw = x2 - x1 + 1.0f, bh = y2 - y1 + 1.0f;
    const float ba = bw * bh;
    float mo = -1e30f; int am = 0;
#pragma unroll 5
    for (int g = 0; g < GMAX; ++g) {
      const float gx1 = s_gt[g * 5 + 0], gy1 = s_gt[g * 5 + 1];
      const float gx2 = s_gt[g * 5 + 2], gy2 = s_gt[g * 5 + 3];
      const float gw = gx2 - gx1 + 1.0f, gh = gy2 - gy1 + 1.0f;
      const float ga = gw * gh;
      float iw = fminf(x2, gx2) - fmaxf(x1, gx1) + 1.0f; iw = fmaxf(iw, 0.0f);
      float ih = fminf(y2, gy2) - fmaxf(y1, gy1) + 1.0f; ih = fmaxf(ih, 0.0f);
      const float inter = iw * ih;
      float ov = inter / (ba + ga - inter);
      if (gw == 1.0f && gh == 1.0f) ov = 0.0f;          // padded gt
      if (ov > mo) { mo = ov; am = g; }                 // first-max semantics
    }
    if (bw == 1.0f && bh == 1.0f) { mo = -1.0f; am = 0; }  // zero box row

    const unsigned bits = rand_bits(kk0, kk1, i);
    const unsigned cls = (mo >= 0.5f) ? 2u : ((mo >= 0.0f) ? 1u : 0u);
    if (cls == 2u) localfg++;
    s_key[i] = (bits & 0xFFFFE000u) | (unsigned)i;       // rand-major, idx tiebreak
    s_ab[i]  = (unsigned char)((cls << 6) | (unsigned)am);
  }
  atomicAdd(&s_nfg, localfg);
  __syncthreads();
  int fgc = s_nfg; if (fgc > FGPI) fgc = FGPI;

  // ---- Phase C: 128 masked min-selections (random-order fg then bg) --
  for (int slot = 0; slot < RPI; ++slot) {
    const unsigned want = (slot < fgc) ? 2u : 1u;
    unsigned m = 0xFFFFFFFFu;
    for (int i = t; i < NTOT; i += 256) {
      if ((unsigned)(s_ab[i] >> 6) == want) {
        const unsigned k = s_key[i];
        m = (k < m) ? k : m;
      }
    }
    s_red[t] = m;
    __syncthreads();
    for (int s = 128; s > 0; s >>= 1) {
      if (t < s) { const unsigned o = s_red[t + s]; if (o < s_red[t]) s_red[t] = o; }
      __syncthreads();
    }
    if (t == 0) {
      const unsigned best = s_red[0];
      const int bi = (best == 0xFFFFFFFFu) ? slot : (int)(best & 0x1FFFu);
      s_keep[slot] = bi;
      s_isfg[slot] = (unsigned char)(slot < fgc);
      s_ab[bi] &= 0x3Fu;   // remove from candidate pool, keep gt_assign bits
    }
    __syncthreads();
  }

  // ---- Phase D: outputs ----------------------------------------------
  const size_t OUT_LAB = (size_t)B * RPI * 5;
  const size_t OUT_BT  = OUT_LAB + (size_t)B * RPI;
  const size_t OUT_BIW = OUT_BT + (size_t)B * RPI * TW;
  float* bt  = out + OUT_BT  + (size_t)img * RPI * TW;
  float* biw = out + OUT_BIW + (size_t)img * RPI * TW;
  for (int k = t; k < RPI * TW; k += 256) { bt[k] = 0.0f; biw[k] = 0.0f; }
  __syncthreads();

  if (t < RPI) {
    const int bi = s_keep[t];
    const int fg = s_isfg[t];
    float x1, y1, x2, y2;
    if (bi < NPROP) {
      const float* rp = all_rois + ((size_t)img * NPROP + bi) * 5;
      x1 = rp[1]; y1 = rp[2]; x2 = rp[3]; y2 = rp[4];
    } else {
      const float* gp = &s_gt[(bi - NPROP) * 5];
      x1 = gp[0]; y1 = gp[1]; x2 = gp[2]; y2 = gp[3];
    }
    const int ga = s_ab[bi] & 31;
    const float gx1 = s_gt[ga * 5 + 0], gy1 = s_gt[ga * 5 + 1];
    const float gx2 = s_gt[ga * 5 + 2], gy2 = s_gt[ga * 5 + 3];
    const float lab = fg ? s_gt[ga * 5 + 4] : 0.0f;

    float* rb = out + ((size_t)img * RPI + t) * 5;
    rb[0] = (float)img; rb[1] = x1; rb[2] = y1; rb[3] = x2; rb[4] = y2;
    out[OUT_LAB + (size_t)img * RPI + t] = lab;

    const float ew = x2 - x1 + 1.0f, eh = y2 - y1 + 1.0f;
    const float ecx = x1 + 0.5f * ew, ecy = y1 + 0.5f * eh;
    const float gw = gx2 - gx1 + 1.0f, gh = gy2 - gy1 + 1.0f;
    const float gcx = gx1 + 0.5f * gw, gcy = gy1 + 0.5f * gh;
    float t0 = (gcx - ecx) / ew * 10.0f;     // (x - mean)/0.1
    float t1 = (gcy - ecy) / eh * 10.0f;
    float t2 = logf(gw / ew) * 5.0f;         // (x - mean)/0.2
    float t3 = logf(gh / eh) * 5.0f;

    const int c = (int)lab;
    if (c > 0) {
      float* btp = bt  + t * TW + 4 * c;
      float* bwp = biw + t * TW + 4 * c;
      btp[0] = t0; btp[1] = t1; btp[2] = t2; btp[3] = t3;
      bwp[0] = 1.0f; bwp[1] = 1.0f; bwp[2] = 1.0f; bwp[3] = 1.0f;
    }
  }
}

extern "C" void kernel_launch(void* const* d_in, const int* in_sizes, int n_in,
                              void* d_out, int out_size, void* d_ws, size_t ws_size,
                              hipStream_t stream) {
  (void)n_in; (void)out_size; (void)d_ws; (void)ws_size;
  const float* all_rois  = (const float*)d_in[0];
  const float* gt_boxes  = (const float*)d_in[1];
  const int*   num_boxes = (const int*)d_in[2];
  float* out = (float*)d_out;
  const int B = in_sizes[0] / (NPROP * 5);
  proposal_target_kernel<<<B, 256, 0, stream>>>(all_rois, gt_boxes, num_boxes, out, B);
}